// _MatchingLayer_46420006536005
// MI455X (gfx1250) — compile-verified
//
#include <hip/hip_runtime.h>
#include <math.h>

typedef __attribute__((ext_vector_type(2))) float v2f;
typedef __attribute__((ext_vector_type(8))) float v8f;

#define B_SZ   64
#define N_SZ   128
#define D_SZ   128
#define R_SZ   4096
#define K_D    64
#define RT     128            // regions per block
#define LDA    132            // padded LDS row stride (floats): 132 % 64 = 4 -> conflict-free

// LDS carve (bytes)
#define PT_OFF 0              // 128 * float4  point  {lat_deg, cos(lat_rad), lon_deg, 0}
#define RG_OFF 2048           // 128 * float4  region {lat_deg, cos(lat_rad), lon_deg, 0}
#define AS_OFF 4096           // 128 x 132 floats : S[b] tile
#define BS_OFF 71680          // 128 x 132 floats : region_emb tile
#define ET_OFF 139264         // 64 floats : E_d_match
#define MK_OFF 139520         // 128 uint  : key padding mask
#define SMEM_BYTES 140032

__global__ __launch_bounds__(256) void matching_fused_kernel(
    const float* __restrict__ S,
    const float* __restrict__ region_emb,
    const float* __restrict__ traj_lat,
    const float* __restrict__ traj_lon,
    const float* __restrict__ region_centroids,
    const unsigned char* __restrict__ key_padding_mask,
    const float* __restrict__ E_d_match,
    float* __restrict__ out)
{
    extern __shared__ __align__(16) char smem[];
    float4*       ptF4  = (float4*)(smem + PT_OFF);
    float4*       rgF4  = (float4*)(smem + RG_OFF);
    float*        As    = (float*)(smem + AS_OFF);
    float*        Bs    = (float*)(smem + BS_OFF);
    float*        Etab  = (float*)(smem + ET_OFF);
    unsigned int* maskU = (unsigned int*)(smem + MK_OFF);

    const int tid   = threadIdx.x;
    const int b     = blockIdx.y;
    const int rbase = blockIdx.x * RT;
    const int lane  = tid & 31;
    const int wave  = tid >> 5;
    const int l15   = lane & 15;
    const int hi    = lane >> 4;        // 0: M=0..7 half, 1: M=8..15 half

    const float deg2rad = 0.017453292519943295f;

    // ---------------- stage to LDS ----------------
    if (tid < 128) {
        float la = traj_lat[b * N_SZ + tid];
        float lo = traj_lon[b * N_SZ + tid];
        ptF4[tid] = make_float4(la, cosf(la * deg2rad), lo, 0.0f);

        float cla = region_centroids[(size_t)(rbase + tid) * 2 + 0];
        float clo = region_centroids[(size_t)(rbase + tid) * 2 + 1];
        rgF4[tid] = make_float4(cla, cosf(cla * deg2rad), clo, 0.0f);

        maskU[tid] = (unsigned int)key_padding_mask[b * N_SZ + tid];
    }
    if (tid < K_D) Etab[tid] = E_d_match[tid];

    for (int i = tid; i < 128 * 32; i += 256) {
        int row = i >> 5;
        int c4  = (i & 31) << 2;
        float4 va = *(const float4*)(S + ((size_t)b * N_SZ + row) * D_SZ + c4);
        *(float4*)(As + row * LDA + c4) = va;
        float4 vb = *(const float4*)(region_emb + (size_t)(rbase + row) * D_SZ + c4);
        *(float4*)(Bs + row * LDA + c4) = vb;
    }
    __syncthreads();

    // ---------------- GEMM: 8 n-tiles x 1 r-column-strip per wave ----------------
    v8f acc[8];
    v8f zero = {0.f, 0.f, 0.f, 0.f, 0.f, 0.f, 0.f, 0.f};
#pragma unroll
    for (int t = 0; t < 8; ++t) acc[t] = zero;

    const int   rcol = (wave << 4) | l15;           // 0..127 within tile
    const float* Bp  = Bs + rcol * LDA + 2 * hi;    // B frag: lane holds K = 2*hi + {0,1}
    const float* Ap  = As + l15  * LDA + 2 * hi;    // A frag: same K split per lane half

    for (int k = 0; k < 128; k += 4) {
        v2f bfrag = *(const v2f*)(Bp + k);
#pragma unroll
        for (int nt = 0; nt < 8; ++nt) {
            v2f afrag = *(const v2f*)(Ap + nt * 16 * LDA + k);
            acc[nt] = __builtin_amdgcn_wmma_f32_16x16x4_f32(
                false, afrag, false, bfrag, (short)0, acc[nt], false, false);
        }
    }

    // ---------------- fuse distance bias + scale + mask into scores ----------------
    const float4 rg     = rgF4[rcol];
    const float  invsq  = 0.08838834764831845f;     // 1/sqrt(128)
    const float  h2r    = 0.008726646259971648f;    // deg2rad / 2
    const float  ninf   = -__builtin_inff();

#pragma unroll
    for (int nt = 0; nt < 8; ++nt) {
#pragma unroll
        for (int j = 0; j < 8; ++j) {
            int n = nt * 16 + 8 * hi + j;           // WMMA C layout: VGPR j, lane-half hi
            float4 pt = ptF4[n];
            // sin(x) = x(1 - x^2/6): |x| < 0.009 rad -> exact to ~1e-12
            float hdlat = (rg.x - pt.x) * h2r;
            float hdlon = (rg.z - pt.z) * h2r;
            float sd = hdlat * (1.0f - hdlat * hdlat * 0.16666667f);
            float se = hdlon * (1.0f - hdlon * hdlon * 0.16666667f);
            float cc = pt.y * rg.y;                 // cos(lat1)cos(lat2)
            float a  = sd * sd + cc * (se * se);
            a = fminf(fmaxf(a, 0.0f), 1.0f);
            // d = 2R asin(sqrt(a)) ~= 2R sqrt(a)(1 + a/6) within clip range
            float d = 12742.0f * sqrtf(a) * (1.0f + 0.16666667f * a);
            float dbin = fminf(fminf(d, 200.0f) * 0.315f, 62.999996f); // *(K_D-1)/D_MAX
            int   kk   = (int)dbin;
            float frac = dbin - (float)kk;
            float lo   = Etab[kk];
            float hv   = Etab[kk + 1];
            float bias = fmaf(frac, hv - lo, lo);
            float s = fmaf(acc[nt][j], invsq, bias);
            if (maskU[n]) s = ninf;
            acc[nt] = acc[nt];                      // keep vector live
            acc[nt][j] = s;
        }
    }

    // ---------------- softmax over n (axis=1) + weighted sum, per region column ----------------
    float m = ninf;
#pragma unroll
    for (int nt = 0; nt < 8; ++nt)
#pragma unroll
        for (int j = 0; j < 8; ++j) m = fmaxf(m, acc[nt][j]);
    m = fmaxf(m, __shfl_xor(m, 16, 32));            // join the two lane halves (same r)

    float sum_e = 0.0f, sum_es = 0.0f;
#pragma unroll
    for (int nt = 0; nt < 8; ++nt) {
#pragma unroll
        for (int j = 0; j < 8; ++j) {
            float s = acc[nt][j];
            if (s != ninf) {                        // masked lanes contribute 0/0-safe
                float e = __expf(s - m);
                sum_e  += e;
                sum_es  = fmaf(e, s, sum_es);
            }
        }
    }
    sum_e  += __shfl_xor(sum_e, 16, 32);
    sum_es += __shfl_xor(sum_es, 16, 32);

    float o = (sum_e > 0.0f) ? (sum_es / sum_e) : 0.0f;  // nan_to_num for all-masked rows
    if (hi == 0) out[(size_t)b * R_SZ + rbase + rcol] = o;
}

extern "C" void kernel_launch(void* const* d_in, const int* in_sizes, int n_in,
                              void* d_out, int out_size, void* d_ws, size_t ws_size,
                              hipStream_t stream) {
    (void)in_sizes; (void)n_in; (void)d_ws; (void)ws_size; (void)out_size;
    const float*         S    = (const float*)d_in[0];
    const float*         emb  = (const float*)d_in[1];
    const float*         tlat = (const float*)d_in[2];
    const float*         tlon = (const float*)d_in[3];
    const float*         cent = (const float*)d_in[4];
    const unsigned char* mask = (const unsigned char*)d_in[5];
    const float*         Etab = (const float*)d_in[6];
    float*               out  = (float*)d_out;

    dim3 grid(R_SZ / RT, B_SZ);   // 32 x 64 = 2048 blocks
    dim3 block(256);              // 8 wave32s
    matching_fused_kernel<<<grid, block, SMEM_BYTES, stream>>>(
        S, emb, tlat, tlon, cent, mask, Etab, out);
}